// QLSTM_65481071406317
// MI455X (gfx1250) — compile-verified
//
#include <hip/hip_runtime.h>
#include <stdint.h>

#define T_STEPS  256
#define BATCH    512
#define DIN      256
#define HID      256
#define NGATE    32      // 4*Q
#define QB       8
#define BM       16      // batch rows per block (one WMMA M-tile)
#define NBLOCKS  (BATCH / BM)   // 32
#define NTHREADS 256            // 8 waves (wave32)

typedef __attribute__((ext_vector_type(2))) float v2f;
typedef __attribute__((ext_vector_type(8))) float v8f;
typedef __attribute__((ext_vector_type(4))) unsigned int v4u;
typedef __attribute__((ext_vector_type(4))) int v4i;
typedef __attribute__((ext_vector_type(8))) int v8i;

#if defined(__HIP_DEVICE_COMPILE__) && __has_builtin(__builtin_amdgcn_tensor_load_to_lds)
#define HAVE_TDM 1
#else
#define HAVE_TDM 0
#endif

static __device__ inline v8f v8zero() {
  v8f v;
  for (int i = 0; i < 8; ++i) v[i] = 0.0f;
  return v;
}

// D = A(16x4 f32) * B(4x16 f32) + C ; emits v_wmma_f32_16x16x4_f32
static __device__ inline v8f wmma_f32(v2f a, v2f b, v8f c) {
  return __builtin_amdgcn_wmma_f32_16x16x4_f32(false, a, false, b, (short)0, c,
                                               false, false);
}

static __device__ inline float sigmoidf_(float x) {
  return 1.0f / (1.0f + __expf(-x));
}

#if HAVE_TDM
// Issue a TDM DMA of one contiguous 16KB X slab (4096 f32, 1-D tile) into LDS.
static __device__ inline void tdm_load_x(const float* gsrc, uint32_t lds_off) {
  const uint64_t ga = (uint64_t)(uintptr_t)gsrc;
  v4u g0;
  g0.x = 1u;                                   // count=1, user desc
  g0.y = lds_off;                              // lds_addr (bytes)
  g0.z = (uint32_t)ga;                         // global_addr[31:0]
  g0.w = (uint32_t)((ga >> 32) & 0x01FFFFFFull) | (2u << 30);  // addr[56:32]|type=2
  v8i g1;
  g1[0] = 0x00020000;       // workgroup_mask=0, data_size=2 (4B)
  g1[1] = 4096 << 16;       // tensor_dim0[15:0] = 4096 at bits[63:48]
  g1[2] = 1 << 16;          // tensor_dim0 hi=0, tensor_dim1=1
  g1[3] = 4096 << 16;       // tensor_dim1 hi=0, tile_dim0 = 4096
  g1[4] = 0;                // tile_dim1=0 (unused), tile_dim2=0
  g1[5] = 4096;             // tensor_dim0_stride[31:0]
  g1[6] = 0;
  g1[7] = 0;
  v4i g2 = {0, 0, 0, 0};
  v4i g3 = {0, 0, 0, 0};
  v8i g4 = {0, 0, 0, 0, 0, 0, 0, 0};   // extra group (6-arg toolchain form)
  __builtin_amdgcn_tensor_load_to_lds(g0, g1, g2, g3, g4, 0);
}
static __device__ inline void tdm_wait() {
#if __has_builtin(__builtin_amdgcn_s_wait_tensorcnt)
  __builtin_amdgcn_s_wait_tensorcnt(0);
#else
  asm volatile("s_wait_tensorcnt 0x0" ::: "memory");
#endif
}
#endif

__global__ __launch_bounds__(NTHREADS)
void qlstm_scan_kernel(const float* __restrict__ inputs,       // (T,B,D)
                       const float* __restrict__ W_proj,       // (D+H, 32)
                       const float* __restrict__ b_proj,       // (32,)
                       const float* __restrict__ theta_gate,   // (8,)
                       const float* __restrict__ W_qproj,      // (8, H)
                       const float* __restrict__ b_qproj,      // (H,)
                       const float* __restrict__ theta_attn,   // (8,)
                       const float* __restrict__ W_attn_proj,  // (H, 8)
                       const float* __restrict__ b_attn_proj,  // (8,)
                       const float* __restrict__ W_attn_back,  // (8, H)
                       const float* __restrict__ b_attn_back,  // (H,)
                       float* __restrict__ out)                // outs | hx | cx
{
  __shared__ float s_x[2][BM * DIN];      // double-buffered X slab (32 KB)
  __shared__ float s_h[BM * HID];         // hidden state / hx_pre (16 KB)
  __shared__ float s_gates[BM * NGATE];   // pre-qgate gate inputs (2 KB)
  __shared__ float s_qg[BM * NGATE];      // qgate outputs (2 KB)
  __shared__ float s_part[8 * 256];       // K-split partial tiles (8 KB)
  __shared__ float s_a[BM * 16];          // attn-proj result, N padded to 16
  __shared__ float s_hq[BM * QB];         // attn qgate output

  const int tid  = threadIdx.x;
  const int w    = tid >> 5;    // wave id 0..7
  const int lane = tid & 31;
  const int lh   = lane >> 4;   // lane half (0/1)
  const int ln   = lane & 15;
  const int nt   = w >> 2;      // phase-1 N tile (0/1)
  const int kc   = w & 3;       // phase-1 K chunk (128 wide)
  const int brow0 = blockIdx.x * BM;

  // ---- loop-invariant B-fragments -> registers ----
  v2f wb[32];                   // W_proj frags for (nt, kc)
#pragma unroll
  for (int kk = 0; kk < 32; ++kk) {
    const int krow = kc * 128 + kk * 4 + 2 * lh;
    wb[kk].x = W_proj[krow * NGATE + nt * 16 + ln];
    wb[kk].y = W_proj[(krow + 1) * NGATE + nt * 16 + ln];
  }
  v2f wq[2][2], wab[2][2];
  float bqv[2], bbv[2];
#pragma unroll
  for (int j = 0; j < 2; ++j) {
    const int n0 = (2 * w + j) * 16;
#pragma unroll
    for (int ks = 0; ks < 2; ++ks) {
      const int krow = ks * 4 + 2 * lh;
      wq[j][ks].x  = W_qproj[krow * HID + n0 + ln];
      wq[j][ks].y  = W_qproj[(krow + 1) * HID + n0 + ln];
      wab[j][ks].x = W_attn_back[krow * HID + n0 + ln];
      wab[j][ks].y = W_attn_back[(krow + 1) * HID + n0 + ln];
    }
    bqv[j] = b_qproj[n0 + ln];
    bbv[j] = b_attn_back[n0 + ln];
  }
  v2f wap[8];                   // W_attn_proj frags (N padded 8->16)
#pragma unroll
  for (int kk = 0; kk < 8; ++kk) {
    const int krow = w * 32 + kk * 4 + 2 * lh;
    wap[kk].x = (ln < QB) ? W_attn_proj[krow * QB + ln] : 0.0f;
    wap[kk].y = (ln < QB) ? W_attn_proj[(krow + 1) * QB + ln] : 0.0f;
  }

  float cxr[2][8];              // cell state lives in registers
#pragma unroll
  for (int j = 0; j < 2; ++j)
#pragma unroll
    for (int r = 0; r < 8; ++r) cxr[j][r] = 0.0f;

  // ---- init: zero hidden state, synchronously stage X_0 ----
  for (int i = tid; i < BM * HID; i += NTHREADS) s_h[i] = 0.0f;
  {
    const float* X0 = inputs + (size_t)brow0 * DIN;
    for (int i = tid; i < BM * DIN / 4; i += NTHREADS) {
      const int m = i >> 6, c4 = (i & 63) * 4;
      *(float4*)&s_x[0][m * DIN + c4] = *(const float4*)(X0 + m * DIN + c4);
    }
  }
  __syncthreads();

  for (int t = 0; t < T_STEPS; ++t) {
    const int buf = t & 1;
    // ---- kick off next step's X slab into the other LDS buffer ----
    if (t + 1 < T_STEPS) {
      const float* Xn = inputs + ((size_t)(t + 1) * BATCH + brow0) * DIN;
#if HAVE_TDM
      if (w == 0) tdm_load_x(Xn, (uint32_t)(uintptr_t)&s_x[buf ^ 1][0]);
#else
      __builtin_prefetch((const char*)Xn + tid * 64, 0, 1);
      for (int i = tid; i < BM * DIN / 4; i += NTHREADS) {
        const int m = i >> 6, c4 = (i & 63) * 4;
        *(float4*)&s_x[buf ^ 1][m * DIN + c4] = *(const float4*)(Xn + m * DIN + c4);
      }
#endif
    }

    // ---- Phase 1: gates = [x_t | h] @ W_proj ; 2 N-tiles x 4 K-chunks ----
    {
      // kc 0,1 read the X slab; kc 2,3 read the h slab (hoisted, branchless)
      const float* abase = (kc < 2) ? (s_x[buf] + ln * DIN)
                                    : (s_h + ln * HID - DIN);
      v8f acc = v8zero();
#pragma unroll
      for (int kk = 0; kk < 32; ++kk) {
        const int kg = kc * 128 + kk * 4;
        v2f a;
        a.x = abase[kg + 2 * lh];
        a.y = abase[kg + 2 * lh + 1];
        acc = wmma_f32(a, wb[kk], acc);
      }
#pragma unroll
      for (int r = 0; r < 8; ++r) {
        const int m = r + 8 * lh;
        s_part[w * 256 + m * 16 + ln] = acc[r];
      }
    }
    __syncthreads();
    for (int o = tid; o < BM * NGATE; o += NTHREADS) {
      const int m = o >> 5, n = o & 31;
      const int nt2 = n >> 4, nn = n & 15;
      float s = b_proj[n];
#pragma unroll
      for (int w2 = 0; w2 < 4; ++w2) s += s_part[(nt2 * 4 + w2) * 256 + m * 16 + nn];
      s_gates[m * NGATE + n] = s;
    }
    __syncthreads();

    // ---- Phase 2: analytic quantum gate (prefix products of cos) ----
    if (tid < BM * 4) {
      const int b = tid >> 2, g = tid & 3;
      float c[QB];
#pragma unroll
      for (int q = 0; q < QB; ++q)
        c[q] = __cosf(s_gates[b * NGATE + g * QB + q] + theta_gate[q]);
      float r0 = c[1];
#pragma unroll
      for (int q = 2; q < QB; ++q) r0 *= c[q];
      s_qg[b * NGATE + g * QB + 0] = r0;              // qubit0: XOR b1..b7
      float run = c[0];
#pragma unroll
      for (int q = 1; q < QB; ++q) { run *= c[q]; s_qg[b * NGATE + g * QB + q] = run; }
    }
    __syncthreads();

    // ---- Phase 3: f/i/g/o = qg @ W_qproj + b ; activations ; cell update ----
#pragma unroll
    for (int j = 0; j < 2; ++j) {
      const int n0 = (2 * w + j) * 16;
      v8f g4[4];
#pragma unroll
      for (int g = 0; g < 4; ++g) {
        v8f acc = v8zero();
#pragma unroll
        for (int ks = 0; ks < 2; ++ks) {
          const int k4 = ks * 4;
          v2f a;
          a.x = s_qg[ln * NGATE + g * QB + k4 + 2 * lh];
          a.y = s_qg[ln * NGATE + g * QB + k4 + 2 * lh + 1];
          acc = wmma_f32(a, wq[j][ks], acc);
        }
        g4[g] = acc;
      }
      const float bq = bqv[j];
#pragma unroll
      for (int r = 0; r < 8; ++r) {
        const int m = r + 8 * lh;
        const int col = n0 + ln;
        const float fv = sigmoidf_(g4[0][r] + bq);
        const float iv = sigmoidf_(g4[1][r] + bq);
        const float gv = tanhf(g4[2][r] + bq);
        const float ov = sigmoidf_(g4[3][r] + bq);
        const float cn = fv * cxr[j][r] + iv * gv;
        cxr[j][r] = cn;
        s_h[m * HID + col] = ov * tanhf(cn);          // hx_pre overwrites old h
      }
    }
    __syncthreads();

    // ---- Phase 4: a = hx_pre @ W_attn_proj (N padded), K split 8 ways ----
    {
      v8f acc = v8zero();
#pragma unroll
      for (int kk = 0; kk < 8; ++kk) {
        const int krow = w * 32 + kk * 4 + 2 * lh;
        v2f a;
        a.x = s_h[ln * HID + krow];
        a.y = s_h[ln * HID + krow + 1];
        acc = wmma_f32(a, wap[kk], acc);
      }
#pragma unroll
      for (int r = 0; r < 8; ++r) {
        const int m = r + 8 * lh;
        s_part[w * 256 + m * 16 + ln] = acc[r];
      }
    }
    __syncthreads();
    {
      const int m = tid >> 4, n = tid & 15;
      float s = (n < QB) ? b_attn_proj[n] : 0.0f;
#pragma unroll
      for (int w2 = 0; w2 < 8; ++w2) s += s_part[w2 * 256 + m * 16 + n];
      s_a[m * 16 + n] = s;
    }
    __syncthreads();

    // ---- Phase 5: attn quantum gate ----
    if (tid < BM) {
      float c[QB];
#pragma unroll
      for (int q = 0; q < QB; ++q) c[q] = __cosf(s_a[tid * 16 + q] + theta_attn[q]);
      float r0 = c[1];
#pragma unroll
      for (int q = 2; q < QB; ++q) r0 *= c[q];
      s_hq[tid * QB + 0] = r0;
      float run = c[0];
#pragma unroll
      for (int q = 1; q < QB; ++q) { run *= c[q]; s_hq[tid * QB + q] = run; }
    }
    __syncthreads();

    // ---- Phase 6: h = hq @ W_attn_back + b ----
#pragma unroll
    for (int j = 0; j < 2; ++j) {
      const int n0 = (2 * w + j) * 16;
      v8f acc = v8zero();
#pragma unroll
      for (int ks = 0; ks < 2; ++ks) {
        const int k4 = ks * 4;
        v2f a;
        a.x = s_hq[ln * QB + k4 + 2 * lh];
        a.y = s_hq[ln * QB + k4 + 2 * lh + 1];
        acc = wmma_f32(a, wab[j][ks], acc);
      }
      const float bb = bbv[j];
#pragma unroll
      for (int r = 0; r < 8; ++r) {
        const int m = r + 8 * lh;
        s_h[m * HID + n0 + ln] = acc[r] + bb;
      }
    }
    __syncthreads();

    // ---- store outs[t] slab, fully coalesced float4 ----
    {
      float* dst = out + ((size_t)t * BATCH + brow0) * HID;
      for (int i = tid; i < BM * HID / 4; i += NTHREADS) {
        const int m = i >> 6, c4 = (i & 63) * 4;
        *(float4*)(dst + m * HID + c4) = *(const float4*)(&s_h[m * HID + c4]);
      }
    }
#if HAVE_TDM
    if (w == 0) tdm_wait();     // next-step X slab landed in LDS
#endif
    __syncthreads();
  }

  // ---- final carry (hx, cx) appended after outs ----
  {
    float* hdst = out + (size_t)T_STEPS * BATCH * HID + (size_t)brow0 * HID;
    float* cdst = out + (size_t)T_STEPS * BATCH * HID + (size_t)BATCH * HID +
                  (size_t)brow0 * HID;
    for (int i = tid; i < BM * HID / 4; i += NTHREADS) {
      const int m = i >> 6, c4 = (i & 63) * 4;
      *(float4*)(hdst + m * HID + c4) = *(const float4*)(&s_h[m * HID + c4]);
    }
#pragma unroll
    for (int j = 0; j < 2; ++j) {
      const int n0 = (2 * w + j) * 16;
#pragma unroll
      for (int r = 0; r < 8; ++r) {
        const int m = r + 8 * lh;
        cdst[m * HID + n0 + ln] = cxr[j][r];
      }
    }
  }
}

extern "C" void kernel_launch(void* const* d_in, const int* in_sizes, int n_in,
                              void* d_out, int out_size, void* d_ws, size_t ws_size,
                              hipStream_t stream) {
  (void)in_sizes; (void)n_in; (void)out_size; (void)d_ws; (void)ws_size;
  const float* inputs      = (const float*)d_in[0];
  const float* W_proj      = (const float*)d_in[1];
  const float* b_proj      = (const float*)d_in[2];
  const float* theta_gate  = (const float*)d_in[3];
  const float* W_qproj     = (const float*)d_in[4];
  const float* b_qproj     = (const float*)d_in[5];
  const float* theta_attn  = (const float*)d_in[6];
  const float* W_attn_proj = (const float*)d_in[7];
  const float* b_attn_proj = (const float*)d_in[8];
  const float* W_attn_back = (const float*)d_in[9];
  const float* b_attn_back = (const float*)d_in[10];
  float* out = (float*)d_out;

  qlstm_scan_kernel<<<NBLOCKS, NTHREADS, 0, stream>>>(
      inputs, W_proj, b_proj, theta_gate, W_qproj, b_qproj, theta_attn,
      W_attn_proj, b_attn_proj, W_attn_back, b_attn_back, out);
}